// Retriever_v3_88244398063935
// MI455X (gfx1250) — compile-verified
//
#include <hip/hip_runtime.h>
#include <math.h>

typedef float v2f __attribute__((ext_vector_type(2)));
typedef float v4f __attribute__((ext_vector_type(4)));
typedef float v8f __attribute__((ext_vector_type(8)));

#define D512 512
#define NB 8
#define NACT 1024
#define KTOP 50
#define PRE_ELEMS 150528           // 3*224*224
#define PRE_V4 (PRE_ELEMS / 4)     // 37632

// ---------------------------------------------------------------------------
// f32 GEMM via V_WMMA_F32_16X16X4_F32. C[M,N] = A[M,K] @ B[K,N], row-major.
// One wave32 per 16x16 output tile; 8 waves per block.
// A frag (16x4): lane l: m = l&15, k-pair = 2*(l>>4); 2 consecutive floats.
// B frag (4x16): lane l: n = l&15, rows kk+2*(l>>4) and +1 (stride-N loads).
// C/D (16x16):   vgpr i: row = i + 8*(l>>4), col = l&15.
// ---------------------------------------------------------------------------
__global__ void wmma_gemm_f32_kernel(const float* __restrict__ A,
                                     const float* __restrict__ B,
                                     float* __restrict__ C,
                                     int M, int N, int K) {
  const int tilesN = N >> 4;
  const int tilesM = (M + 15) >> 4;
  const int tiles = tilesM * tilesN;
  const int wave = blockIdx.x * 8 + (threadIdx.x >> 5);
  if (wave >= tiles) return;                    // wave-uniform exit
  const int tm = wave / tilesN;
  const int tn = wave % tilesN;
  const int lane = threadIdx.x & 31;
  const int half = lane >> 4;                   // 0 or 1
  const int l15 = lane & 15;

  const int rowA = tm * 16 + l15;
  const bool rowOK = rowA < M;
  const int rA = rowOK ? rowA : 0;
  const int colB = tn * 16 + l15;

  v8f acc = {};
  for (int kk = 0; kk < K; kk += 4) {
    v2f a = *(const v2f*)(A + (size_t)rA * K + kk + 2 * half);
    if (!rowOK) { a.x = 0.0f; a.y = 0.0f; }     // lowers to v_cndmask; EXEC stays full
    v2f b;
    const float* bp = B + (size_t)(kk + 2 * half) * N + colB;
    b.x = bp[0];
    b.y = bp[N];
    acc = __builtin_amdgcn_wmma_f32_16x16x4_f32(false, a, false, b,
                                                (short)0, acc, false, false);
  }
#pragma unroll
  for (int i = 0; i < 8; ++i) {
    int row = tm * 16 + i + 8 * half;
    if (row < M) C[(size_t)row * N + tn * 16 + l15] = acc[i];
  }
}

// ---------------------------------------------------------------------------
// Seq self-attention epilogue (per batch): s=QK^T/sqrt(D), softmax, x=sV,
// w=tanh(x@fcw+b), scores[n]=sum_s w_s*actions[b,s,n].  256 threads/block.
// ---------------------------------------------------------------------------
__global__ void seq_attn_kernel(const float* __restrict__ Q,
                                const float* __restrict__ K,
                                const float* __restrict__ V,
                                const float* __restrict__ actions,
                                const float* __restrict__ fcw,
                                const float* __restrict__ fcb,
                                float* __restrict__ scores) {
  const int b = blockIdx.x, tid = threadIdx.x;
  __shared__ float red[256];
  __shared__ float sp[16];
  __shared__ float wv[4];
  const float scale = 1.0f / sqrtf(512.0f);

  // s[i][j]: 16 groups of 16 threads
  const int g = tid >> 4, t = tid & 15, i = g >> 2, j = g & 3;
  const float* qr = Q + (size_t)(b * 4 + i) * D512;
  const float* kr = K + (size_t)(b * 4 + j) * D512;
  float part = 0.0f;
  for (int d = t; d < D512; d += 16) part += qr[d] * kr[d];
  red[tid] = part;
  __syncthreads();
  if (t == 0) {
    float s = 0.0f;
    for (int u = 0; u < 16; ++u) s += red[g * 16 + u];
    sp[g] = s * scale;
  }
  __syncthreads();
  if (tid < 4) {                                  // softmax over row tid
    float m = sp[tid * 4];
    for (int jj = 1; jj < 4; ++jj) m = fmaxf(m, sp[tid * 4 + jj]);
    float e[4], s = 0.0f;
    for (int jj = 0; jj < 4; ++jj) { e[jj] = expf(sp[tid * 4 + jj] - m); s += e[jj]; }
    for (int jj = 0; jj < 4; ++jj) sp[tid * 4 + jj] = e[jj] / s;
  }
  __syncthreads();
  // w[i] = tanh(dot(x_i, fcw)+b): 4 groups of 64 threads
  const int i2 = tid >> 6, t2 = tid & 63;
  float p2 = 0.0f;
  for (int d = t2; d < D512; d += 64) {
    float x = sp[i2 * 4 + 0] * V[(size_t)(b * 4 + 0) * D512 + d]
            + sp[i2 * 4 + 1] * V[(size_t)(b * 4 + 1) * D512 + d]
            + sp[i2 * 4 + 2] * V[(size_t)(b * 4 + 2) * D512 + d]
            + sp[i2 * 4 + 3] * V[(size_t)(b * 4 + 3) * D512 + d];
    p2 += x * fcw[d];
  }
  red[tid] = p2;
  __syncthreads();
  for (int off = 32; off >= 1; off >>= 1) {
    if (t2 < off) red[tid] += red[tid + off];
    __syncthreads();
  }
  if (t2 == 0) wv[i2] = tanhf(red[tid] + fcb[0]);
  __syncthreads();
  for (int n = tid; n < NACT; n += 256) {
    float s = wv[0] * actions[(size_t)(b * 4 + 0) * NACT + n]
            + wv[1] * actions[(size_t)(b * 4 + 1) * NACT + n]
            + wv[2] * actions[(size_t)(b * 4 + 2) * NACT + n]
            + wv[3] * actions[(size_t)(b * 4 + 3) * NACT + n];
    scores[(size_t)b * NACT + n] = s;
  }
}

// per-batch mean/std(ddof=1) then in-place p_db = clip(sigmoid(z))
__global__ void stats_kernel(float* __restrict__ scores) {
  const int b = blockIdx.x, t = threadIdx.x;
  __shared__ float red[256];
  __shared__ float mu, sd;
  float s = 0.0f;
  for (int n = t; n < NACT; n += 256) s += scores[(size_t)b * NACT + n];
  red[t] = s; __syncthreads();
  for (int off = 128; off >= 1; off >>= 1) { if (t < off) red[t] += red[t + off]; __syncthreads(); }
  if (t == 0) mu = red[0] / 1024.0f;
  __syncthreads();
  float m = mu, sq = 0.0f;
  for (int n = t; n < NACT; n += 256) { float d = scores[(size_t)b * NACT + n] - m; sq += d * d; }
  red[t] = sq; __syncthreads();
  for (int off = 128; off >= 1; off >>= 1) { if (t < off) red[t] += red[t + off]; __syncthreads(); }
  if (t == 0) sd = sqrtf(red[0] / 1023.0f);
  __syncthreads();
  float inv = 1.0f / sd;
  for (int n = t; n < NACT; n += 256) {
    float z = (scores[(size_t)b * NACT + n] - m) * inv;
    float p = 1.0f / (1.0f + expf(-z));
    scores[(size_t)b * NACT + n] = fminf(fmaxf(p, 1e-10f), 1.0f - 1e-10f);
  }
}

// top-50 per batch (jax tie-break: equal value -> lower index first)
__global__ void topk_kernel(const float* __restrict__ pdb,
                            const float* __restrict__ data_id,
                            int* __restrict__ topk_idx,
                            float* __restrict__ out_candid) {
  const int b = blockIdx.x, t = threadIdx.x;
  __shared__ float vals[NACT];
  __shared__ float bv[256];
  __shared__ int bi[256];
  for (int n = t; n < NACT; n += 256) vals[n] = pdb[(size_t)b * NACT + n];
  __syncthreads();
  for (int it = 0; it < KTOP; ++it) {
    float best = -INFINITY; int bidx = 0x7fffffff;
    for (int n = t; n < NACT; n += 256) {
      float v = vals[n];
      if (v > best || (v == best && n < bidx)) { best = v; bidx = n; }
    }
    bv[t] = best; bi[t] = bidx;
    __syncthreads();
    for (int off = 128; off >= 1; off >>= 1) {
      if (t < off) {
        float v = bv[t + off]; int ix = bi[t + off];
        if (v > bv[t] || (v == bv[t] && ix < bi[t])) { bv[t] = v; bi[t] = ix; }
      }
      __syncthreads();
    }
    if (t == 0) {
      int w = bi[0];
      topk_idx[b * KTOP + it] = w;
      out_candid[b * KTOP + it] = data_id[w];
      vals[w] = -INFINITY;
    }
    __syncthreads();
  }
}

// candidates = data_emb[idx] : one block (128 thr) per (b,j), float4 copies
__global__ void gather_emb_kernel(const v4f* __restrict__ emb,
                                  const int* __restrict__ idx,
                                  v4f* __restrict__ out_cand) {
  const int pair = blockIdx.x, t = threadIdx.x;
  const int id = idx[pair];
  out_cand[(size_t)pair * 128 + t] = emb[(size_t)id * 128 + t];
}

// candidates_pre = data_pre[idx] : 241 MB stream -> non-temporal float4
__global__ void gather_pre_kernel(const v4f* __restrict__ src,
                                  const int* __restrict__ idx,
                                  v4f* __restrict__ dst) {
  const int pair = blockIdx.y;                     // 0..399
  const int off = blockIdx.x * blockDim.x + threadIdx.x;  // 0..37631
  const int id = idx[pair];
  v4f v = __builtin_nontemporal_load(&src[(size_t)id * PRE_V4 + off]);
  __builtin_nontemporal_store(v, &dst[(size_t)pair * PRE_V4 + off]);
}

// fusion[b] = concat(txt_eb[b] (4 rows), img[b] (1 row))
__global__ void fusion_kernel(const float* __restrict__ txt_eb,
                              const float* __restrict__ img,
                              float* __restrict__ fusion) {
  const int b = blockIdx.x, t = threadIdx.x;
  for (int e = t; e < 5 * D512; e += 256) {
    int r = e >> 9, c = e & 511;
    fusion[(size_t)b * (5 * D512) + e] =
        (r < 4) ? txt_eb[(size_t)(b * 4 + r) * D512 + c] : img[(size_t)b * D512 + c];
  }
}

// cross-attention epilogue per (b,j): s=qK^T/sqrt(D), softmax(5), x2=sV,
// p_logit = x2.fc_p + b, q = tanh(x2.fc_q + b).  256 threads/block.
__global__ void cross_epilogue_kernel(const float* __restrict__ Qc,
                                      const float* __restrict__ Kc,
                                      const float* __restrict__ Vc,
                                      const float* __restrict__ pw,
                                      const float* __restrict__ pb,
                                      const float* __restrict__ qw,
                                      const float* __restrict__ qb,
                                      float* __restrict__ plogit,
                                      float* __restrict__ out_q) {
  const int bj = blockIdx.x;           // 0..399
  const int b = bj / KTOP;
  const int t = threadIdx.x;
  __shared__ float red[256];
  __shared__ float sl[5];
  const float scale = 1.0f / sqrtf(512.0f);
  const float* q = Qc + (size_t)bj * D512;
  const float* Kb = Kc + (size_t)b * 5 * D512;
  const float* Vb = Vc + (size_t)b * 5 * D512;

  for (int l = 0; l < 5; ++l) {
    float part = 0.0f;
    for (int d = t; d < D512; d += 256) part += q[d] * Kb[l * D512 + d];
    red[t] = part; __syncthreads();
    for (int off = 128; off >= 1; off >>= 1) { if (t < off) red[t] += red[t + off]; __syncthreads(); }
    if (t == 0) sl[l] = red[0] * scale;
    __syncthreads();
  }
  if (t == 0) {
    float m = sl[0];
    for (int l = 1; l < 5; ++l) m = fmaxf(m, sl[l]);
    float s = 0.0f, e[5];
    for (int l = 0; l < 5; ++l) { e[l] = expf(sl[l] - m); s += e[l]; }
    for (int l = 0; l < 5; ++l) sl[l] = e[l] / s;
  }
  __syncthreads();
  float pp = 0.0f, pq = 0.0f;
  for (int d = t; d < D512; d += 256) {
    float x2 = sl[0] * Vb[0 * D512 + d] + sl[1] * Vb[1 * D512 + d] +
               sl[2] * Vb[2 * D512 + d] + sl[3] * Vb[3 * D512 + d] +
               sl[4] * Vb[4 * D512 + d];
    pp += x2 * pw[d];
    pq += x2 * qw[d];
  }
  red[t] = pp; __syncthreads();
  for (int off = 128; off >= 1; off >>= 1) { if (t < off) red[t] += red[t + off]; __syncthreads(); }
  if (t == 0) plogit[bj] = red[0] + pb[0];
  __syncthreads();
  red[t] = pq; __syncthreads();
  for (int off = 128; off >= 1; off >>= 1) { if (t < off) red[t] += red[t + off]; __syncthreads(); }
  if (t == 0) out_q[bj] = tanhf(red[0] + qb[0]);
}

// per batch: p = clip(softmax(logits)); argmax (first-occurrence); gathers
__global__ void finalize_kernel(const float* __restrict__ plogit,
                                const float* __restrict__ cand,
                                const float* __restrict__ candid,
                                float* __restrict__ out_p,
                                float* __restrict__ out_maxc,
                                float* __restrict__ out_maxid) {
  const int b = blockIdx.x, t = threadIdx.x;   // 64 threads
  __shared__ float pv[KTOP];
  __shared__ float ms[2];
  __shared__ int am;
  if (t < KTOP) pv[t] = plogit[b * KTOP + t];
  __syncthreads();
  if (t == 0) {
    float m = pv[0];
    for (int j = 1; j < KTOP; ++j) m = fmaxf(m, pv[j]);
    float s = 0.0f;
    for (int j = 0; j < KTOP; ++j) s += expf(pv[j] - m);
    ms[0] = m; ms[1] = s;
  }
  __syncthreads();
  if (t < KTOP) {
    float p = expf(pv[t] - ms[0]) / ms[1];
    p = fminf(fmaxf(p, 1e-10f), 1.0f - 1e-10f);
    pv[t] = p;
    out_p[b * KTOP + t] = p;
  }
  __syncthreads();
  if (t == 0) {
    float best = pv[0]; int bi = 0;
    for (int j = 1; j < KTOP; ++j) if (pv[j] > best) { best = pv[j]; bi = j; }
    am = bi;
    out_maxid[b] = candid[b * KTOP + bi];
  }
  __syncthreads();
  const int bi = am;
  for (int d = t; d < D512; d += 64)
    out_maxc[(size_t)b * D512 + d] = cand[(size_t)(b * KTOP + bi) * D512 + d];
}

// ---------------------------------------------------------------------------
extern "C" void kernel_launch(void* const* d_in, const int* in_sizes, int n_in,
                              void* d_out, int out_size, void* d_ws, size_t ws_size,
                              hipStream_t stream) {
  const float* img        = (const float*)d_in[0];
  const float* txt_eb     = (const float*)d_in[1];
  const float* txt_matric = (const float*)d_in[2];   // [32,512] flat
  const float* actions    = (const float*)d_in[3];
  const float* data_emb   = (const float*)d_in[4];
  const float* data_pre   = (const float*)d_in[5];
  const float* data_id    = (const float*)d_in[6];
  const float* wq_s = (const float*)d_in[7];
  const float* wk_s = (const float*)d_in[8];
  const float* wv_s = (const float*)d_in[9];
  const float* wq_c = (const float*)d_in[10];
  const float* wk_c = (const float*)d_in[11];
  const float* wv_c = (const float*)d_in[12];
  const float* fcs_w = (const float*)d_in[13];
  const float* fcs_b = (const float*)d_in[14];
  const float* fcp_w = (const float*)d_in[15];
  const float* fcp_b = (const float*)d_in[16];
  const float* fcq_w = (const float*)d_in[17];
  const float* fcq_b = (const float*)d_in[18];

  // output layout (flat concat, float32)
  float* out = (float*)d_out;
  float* out_p      = out;                 // 400
  float* out_q      = out + 400;           // 400
  float* out_maxc   = out + 800;           // 4096
  float* out_maxid  = out + 4896;          // 8
  float* out_cand   = out + 4904;          // 204800
  float* out_candid = out + 209704;        // 400
  float* out_pre    = out + 210104;        // 60211200
  float* out_pdb    = out + 60421304;      // 8192

  // workspace layout (floats)
  float* ws    = (float*)d_ws;
  float* wQs   = ws;                 // 32*512
  float* wKs   = ws + 16384;
  float* wVs   = ws + 32768;
  float* wFus  = ws + 49152;         // 40*512
  float* wQc   = ws + 69632;         // 400*512
  float* wKc   = ws + 274432;        // 40*512
  float* wVc   = ws + 294912;
  float* wPl   = ws + 315392;        // 400
  int*   wIdx  = (int*)(ws + 315792);// 400 ints

  auto gemm = [&](const float* A, const float* B, float* C, int M) {
    int tiles = ((M + 15) / 16) * (D512 / 16);
    wmma_gemm_f32_kernel<<<(tiles + 7) / 8, 256, 0, stream>>>(A, B, C, M, D512, D512);
  };

  // 1) seq projections: [32,512] @ [512,512]
  gemm(txt_matric, wq_s, wQs, 32);
  gemm(txt_matric, wk_s, wKs, 32);
  gemm(txt_matric, wv_s, wVs, 32);

  // 2) seq attention + scores (into p_db slot), 3) stats -> p_db in place
  seq_attn_kernel<<<NB, 256, 0, stream>>>(wQs, wKs, wVs, actions, fcs_w, fcs_b, out_pdb);
  stats_kernel<<<NB, 256, 0, stream>>>(out_pdb);

  // 4) top-50 per batch (also writes candidates_id)
  topk_kernel<<<NB, 256, 0, stream>>>(out_pdb, data_id, wIdx, out_candid);

  // 5) gathers
  gather_emb_kernel<<<NB * KTOP, 128, 0, stream>>>((const v4f*)data_emb, wIdx, (v4f*)out_cand);
  {
    dim3 g(PRE_V4 / 256, NB * KTOP);   // 147 x 400
    gather_pre_kernel<<<g, 256, 0, stream>>>((const v4f*)data_pre, wIdx, (v4f*)out_pre);
  }

  // 6) fusion + cross projections
  fusion_kernel<<<NB, 256, 0, stream>>>(txt_eb, img, wFus);
  gemm(out_cand, wq_c, wQc, NB * KTOP);   // [400,512]
  gemm(wFus, wk_c, wKc, NB * 5);          // [40,512]
  gemm(wFus, wv_c, wVc, NB * 5);

  // 7) cross epilogue -> p logits + q, 8) finalize -> p, argmax outputs
  cross_epilogue_kernel<<<NB * KTOP, 256, 0, stream>>>(wQc, wKc, wVc, fcp_w, fcp_b,
                                                       fcq_w, fcq_b, wPl, out_q);
  finalize_kernel<<<NB, 64, 0, stream>>>(wPl, out_cand, out_candid,
                                         out_p, out_maxc, out_maxid);
}